// ScaledDotProductAttention_23330262352474
// MI455X (gfx1250) — compile-verified
//
#include <hip/hip_runtime.h>
#include <math.h>

typedef __attribute__((ext_vector_type(2))) float v2f;
typedef __attribute__((ext_vector_type(4))) float v4f;
typedef __attribute__((ext_vector_type(8))) float v8f;

#define B_   4
#define H_   16
#define S_   2048
#define D_   64
#define QT   16              // query rows per workgroup (one 16-row WMMA tile)
#define NW   4               // waves per workgroup
#define NT   (NW * 32)       // threads per workgroup
#define PADS (S_ + 4)        // padded LDS row stride -> conflict-free strided reads
#define VPAD 68              // padded V-staging / combine row stride

// 4 cooperating wave32s per 16-row query tile. Grid = B*H*(S/16) = 8192 blocks.
__global__ __launch_bounds__(NT)
void ScaledDotProductAttention_23330262352474_kernel(
    const float* __restrict__ q, const float* __restrict__ k,
    const float* __restrict__ v, const int* __restrict__ mask,
    float* __restrict__ ctx_out, float* __restrict__ w_out)
{
    __shared__ float P[QT * PADS];        // 16 x 2048 fp32 exp-scores (~128 KB)
    __shared__ float Scr[NW * QT * VPAD]; // per-wave V staging, then ctx partials (~17 KB)
    __shared__ float red[NT];             // softmax reductions
    __shared__ float invr[QT];            // per-row 1/sum

    const int tid  = threadIdx.x;         // 0..127
    const int wave = tid >> 5;            // 0..3
    const int l16  = tid & 15;            // A/C M-row low bits; B/C N-column
    const int h2   = (tid >> 4) & 1;      // half-wave select

    const int wg = blockIdx.x;
    const int qt = wg & (S_ / QT - 1);
    const int bh = wg / (S_ / QT);        // b*H + h
    const int b  = bh / H_;
    const int q0 = qt * QT;

    const float* qb = q + (size_t)bh * S_ * D_;
    const float* kb = k + (size_t)bh * S_ * D_;
    const float* vb = v + (size_t)bh * S_ * D_;
    const int*   mb = mask + (size_t)b * S_;

    // Per-wave Q fragments for all 16 K-slices (A layout: lane l16 holds row
    // M=l16; VGPR j holds K = d0 + 2*h2 + j). Redundant across waves; L0-cached.
    v2f qa[16];
    #pragma unroll
    for (int t = 0; t < 16; ++t) {
        const float* p = qb + (size_t)(q0 + l16) * D_ + t * 4 + 2 * h2;
        qa[t] = v2f{p[0], p[1]};
    }

    const float scale = 0.125f;           // 1/sqrt(64)

    // ---- Phase 1: scores = (Q K^T)*scale + mask*(-1e10); waves interleave n0 ----
    for (int n0 = wave * 16; n0 < S_; n0 += NW * 16) {
        v8f c = {};
        #pragma unroll
        for (int t = 0; t < 16; ++t) {
            // B-matrix (K^T slice, 4x16): B[kk][n] = K[n0+n][d0+kk]
            const float* p = kb + (size_t)(n0 + l16) * D_ + t * 4 + 2 * h2;
            v2f bv = v2f{p[0], p[1]};
            c = __builtin_amdgcn_wmma_f32_16x16x4_f32(false, qa[t], false, bv,
                                                      (short)0, c, false, false);
        }
        const float mval = (float)mb[n0 + l16] * -1e10f;
        #pragma unroll
        for (int r = 0; r < 8; ++r) {
            const int row = r + 8 * h2;   // C/D layout: M = r + 8*(lane/16)
            P[row * PADS + n0 + l16] = c[r] * scale + mval;
        }
    }
    __syncthreads();

    // ---- Phase 2: row softmax stats (8 lanes per row, float4-vectorized) ----
    {
        const int seg  = tid >> 4;                    // 0..7
        const int base = l16 * PADS + seg * (S_ / 8); // 256 columns per lane
        float m = -INFINITY;
        for (int c = 0; c < S_ / 8; c += 4) {
            const v4f x = *(const v4f*)&P[base + c];
            m = fmaxf(m, fmaxf(fmaxf(x.x, x.y), fmaxf(x.z, x.w)));
        }
        red[tid] = m;
        __syncthreads();
        #pragma unroll
        for (int s = 0; s < 8; ++s) m = fmaxf(m, red[l16 + 16 * s]);
        __syncthreads();

        float sum = 0.0f;
        for (int c = 0; c < S_ / 8; c += 4) {
            const v4f x = *(const v4f*)&P[base + c];
            v4f e;
            e.x = __expf(x.x - m); e.y = __expf(x.y - m);
            e.z = __expf(x.z - m); e.w = __expf(x.w - m);
            *(v4f*)&P[base + c] = e;                  // keep UNNORMALIZED exp in LDS
            sum += e.x + e.y + e.z + e.w;
        }
        red[tid] = sum;
        __syncthreads();
        if (tid < QT) {
            float t = 0.0f;
            #pragma unroll
            for (int s = 0; s < 8; ++s) t += red[tid + 16 * s];
            invr[tid] = 1.0f / t;
        }
        __syncthreads();
    }

    // ---- Phase 2b: attention weights = exp * (1/sum), coalesced float4 ----
    {
        float* wrow = w_out + ((size_t)bh * S_ + q0) * S_;
        for (int r = 0; r < QT; ++r) {
            const float iv = invr[r];
            const v4f* src = (const v4f*)&P[r * PADS];
            v4f*       dst = (v4f*)(wrow + (size_t)r * S_);
            for (int c4 = tid; c4 < S_ / 4; c4 += NT)
                dst[c4] = src[c4] * iv;
        }
    }

    // ---- Phase 3: context = (exp P) @ V; waves interleave 16-deep K chunks ----
    float* Vs = &Scr[wave * QT * VPAD];   // this wave's staging tile
    v8f acc[4] = {v8f{}, v8f{}, v8f{}, v8f{}};
    for (int k0 = wave * 16; k0 < S_; k0 += NW * 16) {
        __syncthreads();                  // all waves: prior Vs reads complete
        const int lane = tid & 31;
        #pragma unroll
        for (int i = 0; i < 8; ++i) {     // stage V[k0..k0+15][0..63], float4
            const int idx = i * 32 + lane;
            const int row = idx >> 4;
            const int c4  = idx & 15;
            const v4f val = *(const v4f*)(vb + (size_t)(k0 + row) * D_ + c4 * 4);
            *(v4f*)&Vs[row * VPAD + c4 * 4] = val;
        }
        __syncthreads();
        #pragma unroll
        for (int kk = 0; kk < 4; ++kk) {
            // A from LDS: row M=l16, K = k0 + kk*4 + 2*h2 + j
            const v2f a = *(const v2f*)&P[l16 * PADS + k0 + kk * 4 + 2 * h2];
            const int r0 = kk * 4 + 2 * h2;
            #pragma unroll
            for (int t = 0; t < 4; ++t) { // 4 independent output tiles
                const v2f bv = v2f{Vs[r0 * VPAD + t * 16 + l16],
                                   Vs[(r0 + 1) * VPAD + t * 16 + l16]};
                acc[t] = __builtin_amdgcn_wmma_f32_16x16x4_f32(
                             false, a, false, bv, (short)0, acc[t], false, false);
            }
        }
    }

    // Reuse Scr as per-wave partial-context buffers, then tree-combine.
    __syncthreads();
    #pragma unroll
    for (int t = 0; t < 4; ++t) {
        #pragma unroll
        for (int r = 0; r < 8; ++r)
            Vs[(r + 8 * h2) * VPAD + t * 16 + l16] = acc[t][r];
    }
    __syncthreads();
    for (int i = tid; i < QT * D_; i += NT) {
        const int row = i >> 6;
        const int col = i & 63;
        float s = 0.0f;
        #pragma unroll
        for (int w = 0; w < NW; ++w)
            s += Scr[w * QT * VPAD + row * VPAD + col];
        ctx_out[((size_t)bh * S_ + q0 + row) * D_ + col] = s * invr[row];
    }
}

extern "C" void kernel_launch(void* const* d_in, const int* in_sizes, int n_in,
                              void* d_out, int out_size, void* d_ws, size_t ws_size,
                              hipStream_t stream) {
    (void)in_sizes; (void)n_in; (void)out_size; (void)d_ws; (void)ws_size;
    const float* q    = (const float*)d_in[0];
    const float* k    = (const float*)d_in[1];
    const float* v    = (const float*)d_in[2];
    const int*   mask = (const int*)d_in[3];

    float* ctx = (float*)d_out;                                  // (B,H,S,D) first
    float* w   = ctx + (size_t)B_ * H_ * S_ * D_;                // then (B,H,S,S)

    dim3 grid(B_ * H_ * (S_ / QT));
    dim3 block(NT);
    ScaledDotProductAttention_23330262352474_kernel<<<grid, block, 0, stream>>>(
        q, k, v, mask, ctx, w);
}